// LinearChainCRF_89524298318194
// MI455X (gfx1250) — compile-verified
//
#include <hip/hip_runtime.h>
#include <stdint.h>

#define AS1 __attribute__((address_space(1)))
#define AS3 __attribute__((address_space(3)))

namespace crf {
constexpr int   Bsz = 256;
constexpr int   P   = 128;
constexpr int   L   = 512;
constexpr int   TCH = 16;          // timesteps per double-buffered chunk
constexpr int   NCH = L / TCH;     // 32 chunks
constexpr int   STR = TCH + 1;     // 17: LDS row stride pad -> conflict-free column reads
constexpr int   G   = 2;           // batch elements per wave (ILP latency hiding)
constexpr float ETA_F = 4.0f;
constexpr float NEGL  = -10000.0f;
constexpr float K2 = 0.0183156393f;     // exp(-ETA)
constexpr float K1 = 1.0f - K2;         // 1 - exp(-ETA)
}

typedef int v4i __attribute__((ext_vector_type(4)));

// ---- gfx1250 async global->LDS copy (ASYNCcnt path) ------------------------
__device__ __forceinline__ void async_b128(const float* gp, float* lp) {
#if __has_builtin(__builtin_amdgcn_global_load_async_to_lds_b128)
  AS1 float* g1 = (AS1 float*)const_cast<float*>(gp);  // addrspacecast
  AS3 float* l3 = (AS3 float*)lp;                      // addrspacecast
  __builtin_amdgcn_global_load_async_to_lds_b128(
      (AS1 v4i*)g1, (AS3 v4i*)l3, 0, 0);
#else
  unsigned lds = (unsigned)(size_t)lp;                 // low 32 bits = LDS offset
  unsigned long long ga = (unsigned long long)(size_t)gp;
  asm volatile("global_load_async_to_lds_b128 %0, %1, off"
               :: "v"(lds), "v"(ga) : "memory");
#endif
}

#if __has_builtin(__builtin_amdgcn_s_wait_asynccnt)
#define WAIT_ASYNC(n) __builtin_amdgcn_s_wait_asynccnt(n)
#else
#define WAIT_ASYNC(n) asm volatile("s_wait_asynccnt " #n ::: "memory")
#endif

// ---- wave32 butterfly sum ---------------------------------------------------
template <int IMM>
__device__ __forceinline__ float swz(float x) {
  return __int_as_float(__builtin_amdgcn_ds_swizzle(__float_as_int(x), IMM));
}
constexpr int dpp8sel(int a, int b, int c, int d, int e, int f, int g, int h) {
  return a | (b << 3) | (c << 6) | (d << 9) | (e << 12) | (f << 15) |
         (g << 18) | (h << 21);
}

__device__ __forceinline__ float wave_sum(float x) {
#if __has_builtin(__builtin_amdgcn_mov_dpp8) && \
    __has_builtin(__builtin_amdgcn_update_dpp) && \
    __has_builtin(__builtin_amdgcn_permlanex16)
  // VALU-only butterfly: no DS round-trips / s_wait_dscnt in the chain.
  int v;
  v = __float_as_int(x);   // xor 1
  x += __int_as_float(__builtin_amdgcn_mov_dpp8(v, dpp8sel(1,0,3,2,5,4,7,6)));
  v = __float_as_int(x);   // xor 2
  x += __int_as_float(__builtin_amdgcn_mov_dpp8(v, dpp8sel(2,3,0,1,6,7,4,5)));
  v = __float_as_int(x);   // xor 4
  x += __int_as_float(__builtin_amdgcn_mov_dpp8(v, dpp8sel(4,5,6,7,0,1,2,3)));
  v = __float_as_int(x);   // xor 8 == row_ror:8 within each 16-lane row
  x += __int_as_float(__builtin_amdgcn_update_dpp(v, v, 0x128, 0xf, 0xf, false));
  v = __float_as_int(x);   // xor 16: identity-mapped cross-half permute
  x += __int_as_float(__builtin_amdgcn_permlanex16(v, v, 0x76543210, 0xfedcba98,
                                                   false, false));
  return x;
#else
  x += swz<0x401f>(x);   // xor 16
  x += swz<0x201f>(x);   // xor 8
  x += swz<0x101f>(x);   // xor 4
  x += swz<0x081f>(x);   // xor 2
  x += swz<0x041f>(x);   // xor 1
  return x;
#endif
}

__device__ __forceinline__ float fast_rcp(float x) {
#if __has_builtin(__builtin_amdgcn_rcpf)
  return __builtin_amdgcn_rcpf(x);
#else
  return 1.0f / x;
#endif
}

// ---- main: one wave handles G=2 batch elements -----------------------------
// Scaled forward algorithm (exp domain, renormalized each step):
//   mm[p] = K1*q[p] + K2   (q = normalized alpha probabilities, sum q = 1)
//   w[p]  = exp(theta[p,t]) * mm[p]        (exp(theta) is OFF the serial chain)
//   S     = sum_p w[p]                     (one VALU butterfly per step)
//   q <- w * rcp(S) ; mm <- fma(K1,q,K2) ; C += log(S)   (log off-chain)
// Init: mm[p] = endpoint_mask[0,p] ? 1 : 0  (exact exp(NEG_LARGE) ~ 0)
// Final: logZ = C. Identical math to the reference logsumexp recursion.
__global__ __launch_bounds__(32) void crf_fwd_kernel(
    const float* __restrict__ theta, const long long* __restrict__ y,
    const unsigned char* __restrict__ emask, float* __restrict__ per_b) {
  using namespace crf;
  __shared__ float smem[G][2][P * STR];   // ~35 KB total, double buffered
  const int lane = threadIdx.x;
  const int b0   = blockIdx.x * G;

  const float* tb0 = theta + (size_t)b0 * P * L;
  const float* tb1 = theta + (size_t)(b0 + 1) * P * L;

  // async-copy geometry: 32 lanes x 16B cover 8 rows x 16 timesteps per issue
  const int prow = lane >> 2;            // 0..7
  const int tcol = (lane & 3) << 2;      // 0,4,8,12

  auto issue = [&](int k, int buf) {
    const int t0 = k * TCH;
    const float* g0 = tb0 + prow * L + t0 + tcol;
    const float* g1 = tb1 + prow * L + t0 + tcol;
    float* l0 = &smem[0][buf][prow * STR + tcol];
    float* l1 = &smem[1][buf][prow * STR + tcol];
    #pragma unroll
    for (int i = 0; i < P / 8; ++i) {    // 16 issues per g -> 32 per chunk
      async_b128(g0, l0);
      async_b128(g1, l1);
      g0 += 8 * L; g1 += 8 * L; l0 += 8 * STR; l1 += 8 * STR;
    }
  };

  // endpoint handling: t=0 folded into mm init; t=L-1 as additive adjustment
  float adjL[4], mm[G][4];
  #pragma unroll
  for (int c = 0; c < 4; ++c) {
    const int p = lane + 32 * c;
    const float m0 = emask[p] ? 1.0f : 0.0f;
    mm[0][c] = m0;
    mm[1][c] = m0;
    adjL[c] = emask[(size_t)(L - 1) * P + p] ? 0.0f : NEGL;
  }

  issue(0, 0);
  issue(1, 1);

  float Cacc[G] = {};                     // running sum of log(S_t) (uniform)
  float acc = 0.0f;                       // per-lane score partial
  const int sg = lane >> 4;               // score duty: batch element
  const int st = lane & 15;               // score duty: local timestep
  const long long* yb = y + (size_t)(b0 + sg) * L;

  for (int k = 0; k < NCH; ++k) {
    // chunk k ready when at most chunk k+1's 32 issues remain (in-order done)
    if (k + 1 < NCH) { WAIT_ASYNC(32); } else { WAIT_ASYNC(0); }
    const int buf = k & 1;
    const float* s0 = smem[0][buf];
    const float* s1 = smem[1][buf];

    for (int tt = 0; tt < TCH; ++tt) {
      const int t = k * TCH + tt;
      float th[G][4];
      #pragma unroll
      for (int c = 0; c < 4; ++c) {       // conflict-free: bank = (17*lane+..)%64
        th[0][c] = s0[(lane + 32 * c) * STR + tt];
        th[1][c] = s1[(lane + 32 * c) * STR + tt];
      }
      if (t == L - 1) {
        #pragma unroll
        for (int c = 0; c < 4; ++c) { th[0][c] += adjL[c]; th[1][c] += adjL[c]; }
      }

      float w[G][4], S[G];
      #pragma unroll
      for (int g = 0; g < G; ++g) {
        #pragma unroll
        for (int c = 0; c < 4; ++c) w[g][c] = __expf(th[g][c]) * mm[g][c];
        const float s4 = (w[g][0] + w[g][1]) + (w[g][2] + w[g][3]);
        S[g] = wave_sum(s4);              // the only cross-lane op in the chain
      }
      #pragma unroll
      for (int g = 0; g < G; ++g) {
        const float inv = fast_rcp(S[g]);
        Cacc[g] += __logf(S[g]);          // off the critical path
        #pragma unroll
        for (int c = 0; c < 4; ++c) mm[g][c] = fmaf(K1, w[g][c] * inv, K2);
      }
    }

    // score (unary gather + transition count) from the staged LDS tile
    {
      const int t  = k * TCH + st;
      const int yt = (int)yb[t];
      const float* sb = (sg == 0) ? s0 : s1;
      float uu = sb[yt * STR + st];
      if (t == 0)     uu += emask[yt] ? 0.0f : NEGL;
      if (t == L - 1) uu += emask[(size_t)(L - 1) * P + yt] ? 0.0f : NEGL;
      acc += uu;
      if (t < L - 1) {
        const int yn = (int)yb[t + 1];
        if (yn != yt) acc -= ETA_F;
      }
    }

    if (k + 2 < NCH) issue(k + 2, buf);   // refill consumed buffer
  }

  // logZ[g] == Cacc[g]; reduce score partials per batch element
  const float sc0 = wave_sum((lane < 16) ? acc : 0.0f);
  const float sc1 = wave_sum((lane >= 16) ? acc : 0.0f);
  if (lane == 0) {
    per_b[b0 + 0] = Cacc[0] - sc0;
    per_b[b0 + 1] = Cacc[1] - sc1;
  }
}

// ---- deterministic mean over 256 per-b values via v_wmma_f32_16x16x4_f32 ---
typedef __attribute__((ext_vector_type(2))) float v2f;
typedef __attribute__((ext_vector_type(8))) float v8f;

__global__ __launch_bounds__(32) void crf_reduce_kernel(const float* __restrict__ v,
                                                        float* __restrict__ out) {
  const int lane = threadIdx.x;
#if __has_builtin(__builtin_amdgcn_wmma_f32_16x16x4_f32)
  // D = A(16x4) * ones(4x16) + C ; sum(all D entries) = 16 * sum(A entries),
  // independent of the exact lane->element mapping.
  v2f ones; ones[0] = 1.0f; ones[1] = 1.0f;
  v8f acc = {};
  #pragma unroll
  for (int c = 0; c < 4; ++c) {
    v2f a;
    a[0] = v[c * 64 + lane];
    a[1] = v[c * 64 + 32 + lane];
    acc = __builtin_amdgcn_wmma_f32_16x16x4_f32(false, a, false, ones,
                                                (short)0, acc, false, false);
  }
  float s = 0.0f;
  #pragma unroll
  for (int i = 0; i < 8; ++i) s += acc[i];
  s = wave_sum(s) * (1.0f / 16.0f);
#else
  float s = 0.0f;
  #pragma unroll
  for (int c = 0; c < 8; ++c) s += v[c * 32 + lane];
  s = wave_sum(s);
#endif
  if (lane == 0) out[0] = s * (1.0f / 256.0f);
}

extern "C" void kernel_launch(void* const* d_in, const int* in_sizes, int n_in,
                              void* d_out, int out_size, void* d_ws, size_t ws_size,
                              hipStream_t stream) {
  (void)in_sizes; (void)n_in; (void)out_size; (void)ws_size;
  const float*          theta = (const float*)d_in[0];
  const long long*      y     = (const long long*)d_in[1];
  const unsigned char*  emask = (const unsigned char*)d_in[2];
  float* per_b = (float*)d_ws;                         // 256 floats scratch

  crf_fwd_kernel<<<crf::Bsz / crf::G, 32, 0, stream>>>(theta, y, emask, per_b);
  crf_reduce_kernel<<<1, 32, 0, stream>>>(per_b, (float*)d_out);
}